// NearAggregator_16767552324011
// MI455X (gfx1250) — compile-verified
//
#include <hip/hip_runtime.h>
#include <math.h>

// Problem constants (reference: B=8192, N=128, D_EMB=128, D_Q=64, fan_in=132)
#define NNB   128
#define DEMB  128
#define DQ    64
#define FANIN 132
#define ROWP  132   // padded LDS row stride (floats): rows stay 16B-aligned for B128,
                    // 132 mod 64 = 4 -> only 2-way bank skew on row-wise pass

typedef __attribute__((ext_vector_type(2))) float v2f;
typedef __attribute__((ext_vector_type(8))) float v8f;

#if __has_builtin(__builtin_amdgcn_wmma_f32_16x16x4_f32)
#define HAVE_WMMA_F32X4 1
#endif
#if __has_builtin(__builtin_amdgcn_global_load_async_to_lds_b128)
#define HAVE_ASYNC_LDS 1
#endif

#ifdef HAVE_ASYNC_LDS
// Parameter types per hipcc diagnostic: arg0 = 'int __attribute__((vector_size(16))) __device__ *'
typedef int v4i __attribute__((vector_size(4 * sizeof(int))));
typedef __attribute__((address_space(1))) v4i* gv4i_p;   // global (device) src
typedef __attribute__((address_space(3))) v4i* lv4i_p;   // LDS dst
static __device__ __forceinline__ void async_wait0() {
#if __has_builtin(__builtin_amdgcn_s_wait_asynccnt)
  __builtin_amdgcn_s_wait_asynccnt(0);
#else
  asm volatile("s_wait_asynccnt 0" ::: "memory");
#endif
}
#endif

// ---------------------------------------------------------------------------
// Kernel 1: q[b, 0:132] = W_key[132,64] @ B_query[b, 0:64]
// One wave per 16 batch rows. M=16 (batch), N=16 (fan_in tile), K=4 per WMMA.
// A fragments live in registers for the whole kernel; B fragments are
// double-buffered so the next tile's 16 loads overlap the current tile's
// 16 back-to-back WMMAs (one wait per tile instead of one per WMMA).
// ---------------------------------------------------------------------------
__global__ __launch_bounds__(32) void qproj_kernel(const float* __restrict__ Bq,
                                                   const float* __restrict__ Wk,
                                                   float* __restrict__ Q) {
  const int b0 = blockIdx.x * 16;
  const int l  = threadIdx.x;     // 0..31
  const int lm = l & 15;
  const int kh = (l >> 4) * 2;    // lanes 0-15 -> K={0,1}; lanes 16-31 -> K={2,3}

#ifdef HAVE_WMMA_F32X4
  // A fragments: B_query rows b0..b0+15, all 16 K-steps.
  v2f a[16];
#pragma unroll
  for (int kk = 0; kk < 16; ++kk) {
    const float* p = Bq + (size_t)(b0 + lm) * DQ + kk * 4 + kh;
    a[kk].x = p[0];
    a[kk].y = p[1];
  }

  v2f bf[2][16];                   // double-buffered B fragments
  // Preload tile 0 (cols 0..15, all valid).
#pragma unroll
  for (int kk = 0; kk < 16; ++kk) {
    const float* p = Wk + (size_t)lm * DQ + kk * 4 + kh;
    bf[0][kk].x = p[0];
    bf[0][kk].y = p[1];
  }

#pragma unroll
  for (int ft = 0; ft < 8; ++ft) {          // full tiles 0..7 (cols < 128)
    const int cur = ft & 1;
    const int nxt = cur ^ 1;
    // Prefetch next tile's B fragments (tile 8 is partial -> guarded loads).
    if (ft < 7) {
#pragma unroll
      for (int kk = 0; kk < 16; ++kk) {
        const float* p = Wk + (size_t)((ft + 1) * 16 + lm) * DQ + kk * 4 + kh;
        bf[nxt][kk].x = p[0];
        bf[nxt][kk].y = p[1];
      }
    } else {
      const int  col = 128 + lm;
      const bool ok  = col < FANIN;
#pragma unroll
      for (int kk = 0; kk < 16; ++kk) {
        bf[nxt][kk].x = ok ? Wk[(size_t)col * DQ + kk * 4 + kh]     : 0.f;
        bf[nxt][kk].y = ok ? Wk[(size_t)col * DQ + kk * 4 + kh + 1] : 0.f;
      }
    }
    // 16 chained WMMAs for the current tile.
    v8f c = {0.f, 0.f, 0.f, 0.f, 0.f, 0.f, 0.f, 0.f};
#pragma unroll
    for (int kk = 0; kk < 16; ++kk)
      c = __builtin_amdgcn_wmma_f32_16x16x4_f32(false, a[kk], false, bf[cur][kk],
                                                (short)0, c, false, false);
    const int col = ft * 16 + lm;
#pragma unroll
    for (int r = 0; r < 8; ++r) {
      const int row = (l < 16) ? r : (r + 8);             // C/D VGPR layout
      Q[(size_t)(b0 + row) * FANIN + col] = c[r];
    }
  }

  // Epilogue: partial tile 8 (cols 128..131 valid) — fragments sit in bf[0].
  {
    v8f c = {0.f, 0.f, 0.f, 0.f, 0.f, 0.f, 0.f, 0.f};
#pragma unroll
    for (int kk = 0; kk < 16; ++kk)
      c = __builtin_amdgcn_wmma_f32_16x16x4_f32(false, a[kk], false, bf[0][kk],
                                                (short)0, c, false, false);
    const int col = 128 + lm;
    if (col < FANIN) {
#pragma unroll
      for (int r = 0; r < 8; ++r) {
        const int row = (l < 16) ? r : (r + 8);
        Q[(size_t)(b0 + row) * FANIN + col] = c[r];
      }
    }
  }
#else
  // Scalar fallback (should not trigger on gfx1250 toolchains)
  for (int idx = l; idx < 16 * FANIN; idx += 32) {
    const int row = idx / FANIN, col = idx % FANIN;
    float acc = 0.f;
    for (int d = 0; d < DQ; ++d)
      acc = fmaf(Wk[(size_t)col * DQ + d], Bq[(size_t)(b0 + row) * DQ + d], acc);
    Q[(size_t)(b0 + row) * FANIN + col] = acc;
  }
#endif
}

// ---------------------------------------------------------------------------
// Kernel 2: one block per batch row b. Stage near_emb[b] (64 KB) in LDS once
// via async global->LDS b128 copies, use it for both the score dot-product
// and the softmax-weighted pooling.
// b_key is dropped: softmax is invariant to a per-row additive constant.
// ---------------------------------------------------------------------------
__global__ __launch_bounds__(128) void agg_kernel(const float* __restrict__ emb,
                                                  const float* __restrict__ dxy,
                                                  const float* __restrict__ dcs,
                                                  const float* __restrict__ Q,
                                                  float* __restrict__ out) {
  __shared__ float E[NNB * ROWP];   // 128 x 132 floats = 67.5 KB
  __shared__ float qs[FANIN];
  __shared__ float sw[NNB];

  const int b = blockIdx.x;
  const int t = threadIdx.x;        // 0..127
  const float* src = emb + (size_t)b * NNB * DEMB;

  qs[t] = Q[(size_t)b * FANIN + t];
  if (t < FANIN - DEMB) qs[DEMB + t] = Q[(size_t)b * FANIN + DEMB + t];

  // ---- stage 128x128 f32 tile -> LDS (padded rows), 16B per lane-op ----
#ifdef HAVE_ASYNC_LDS
#pragma unroll
  for (int i = 0; i < (NNB * DEMB / 4) / 128; ++i) {   // 32 async ops/thread
    const int c   = t + i * 128;
    const int row = c >> 5;
    const int seg = c & 31;
    __builtin_amdgcn_global_load_async_to_lds_b128(
        (gv4i_p)(src + (size_t)row * DEMB + seg * 4),
        (lv4i_p)(&E[row * ROWP + seg * 4]), 0, 0);
  }
  async_wait0();
#else
#pragma unroll
  for (int i = 0; i < (NNB * DEMB / 4) / 128; ++i) {
    const int c   = t + i * 128;
    const int row = c >> 5;
    const int seg = c & 31;
    const float4 v = *(const float4*)(src + (size_t)row * DEMB + seg * 4);
    *(float4*)(&E[row * ROWP + seg * 4]) = v;          // rows 16B-aligned (ROWP=132)
  }
#endif
  __syncthreads();

  // ---- scores: s[t] = E[t,:] . q[0:128] + deltas . q[128:132], scaled ----
  float s = 0.f;
#pragma unroll 8
  for (int f = 0; f < DEMB; ++f) s = fmaf(E[t * ROWP + f], qs[f], s);
  {
    const float2 xy = ((const float2*)dxy)[(size_t)b * NNB + t];
    const float2 cs = ((const float2*)dcs)[(size_t)b * NNB + t];
    s = fmaf(xy.x, qs[128], s);
    s = fmaf(xy.y, qs[129], s);
    s = fmaf(cs.x, qs[130], s);
    s = fmaf(cs.y, qs[131], s);
  }
  s *= 0.125f;                      // 1/sqrt(D_Q)
  sw[t] = s;
  __syncthreads();

  // ---- softmax over the 128 neighbors (LDS broadcast reads) ----
  float mx = -1e30f;
  for (int n = 0; n < NNB; ++n) mx = fmaxf(mx, sw[n]);
  const float e = __expf(s - mx);
  __syncthreads();                  // all reads of scores done before overwrite
  sw[t] = e;
  __syncthreads();
  float sum = 0.f;
  for (int n = 0; n < NNB; ++n) sum += sw[n];
  const float inv = 1.0f / sum;

  // ---- weighted pooling: out[b,t] = sum_n w[n] * E[n,t] (conflict-free) ----
  float acc = 0.f;
#pragma unroll 8
  for (int n = 0; n < NNB; ++n) acc = fmaf(sw[n], E[n * ROWP + t], acc);
  out[(size_t)b * DEMB + t] = acc * inv;
}

// ---------------------------------------------------------------------------
extern "C" void kernel_launch(void* const* d_in, const int* in_sizes, int n_in,
                              void* d_out, int out_size, void* d_ws, size_t ws_size,
                              hipStream_t stream) {
  (void)n_in; (void)out_size; (void)ws_size;
  const float* near_emb = (const float*)d_in[0];
  const float* delta_xy = (const float*)d_in[1];
  const float* delta_cs = (const float*)d_in[2];
  const float* B_query  = (const float*)d_in[3];
  const float* W_key    = (const float*)d_in[4];
  // d_in[5] (b_key) is provably unused: softmax is shift-invariant per row.

  const int Bn = in_sizes[3] / DQ;        // 8192
  float* Qws = (float*)d_ws;              // Bn * 132 floats = 4.3 MB scratch
  float* out = (float*)d_out;

  qproj_kernel<<<Bn / 16, 32, 0, stream>>>(B_query, W_key, Qws);
  agg_kernel<<<Bn, 128, 0, stream>>>(near_emb, delta_xy, delta_cs, Qws, out);
}